// HaarTransform_32177894982114
// MI455X (gfx1250) — compile-verified
//
#include <hip/hip_runtime.h>

typedef __attribute__((ext_vector_type(2))) float v2f;
typedef __attribute__((ext_vector_type(8))) float v8f;
typedef __attribute__((ext_vector_type(4))) int   v4i;

#define LD_IN 68   // 64 + 4 pad: float4-aligned, conflict-free column reads
#define LD_L1 36
#define LD_L2 20
#define LD_T  20   // per-wave 16x16 transpose scratch stride

// One 16x4 (A-layout) / 4x16 (B-layout) slice of the 16x16 Haar operator.
__device__ __forceinline__ v2f haar_op(int k, int lane, float s) {
  const int m    = lane & 15;
  const int cadd = (lane >= 16) ? 2 : 0;
  v2f r;
#pragma unroll
  for (int v = 0; v < 2; ++v) {
    const int j = 4 * k + v + cadd;
    float val = 0.0f;
    if (m == (j >> 1))          val = s;
    else if (m == 8 + (j >> 1)) val = (j & 1) ? -s : s;
    r[v] = val;
  }
  return r;
}

struct WaveCtx {
  int lane, wave, col, rsel, radd, band, cc;
  v2f hA[4], hB[4];
  float* tscr;
};

// One DWT level: N x N input tile in LDS -> four (N/2)^2 bands to global,
// cA also into next-level LDS buffer. All strides/widths compile-time.
template <int N, int W, int LDIN, int LDOUT, bool HAS_NEXT>
__device__ __forceinline__ void haar_level(const float* __restrict__ curIn,
                                           float* __restrict__ nA,
                                           float* __restrict__ outL,
                                           int b, int tX, int tY,
                                           const WaveCtx& c) {
  constexpr int NSUB = N / 16;
  const int oy0 = tY * (N / 2);
  const int ox0 = tX * (N / 2);

  for (int t = c.wave; t < NSUB * NSUB; t += 8) {
    const int tr = t / NSUB, tc = t % NSUB;
    const float* xt = curIn + (tr * 16) * LDIN + tc * 16;

    // ---- stage 1: T = Hv * X (four chained K=4 f32 WMMAs) ----
    v8f acc = {};
#pragma unroll
    for (int k = 0; k < 4; ++k) {
      v2f bx;
      bx[0] = xt[(4 * k + c.rsel + 0) * LDIN + c.col];
      bx[1] = xt[(4 * k + c.rsel + 1) * LDIN + c.col];
      acc = __builtin_amdgcn_wmma_f32_16x16x4_f32(
          false, c.hA[k], false, bx, (short)0, acc, false, false);
    }

    // ---- D-layout -> A-layout via per-wave LDS scratch ----
#pragma unroll
    for (int r = 0; r < 8; ++r)
      c.tscr[(r + c.radd) * LD_T + c.col] = acc[r];
    asm volatile("s_wait_dscnt 0" ::: "memory");

    // ---- stage 2: Y = T * Hh^T ----
    v8f y = {};
#pragma unroll
    for (int k = 0; k < 4; ++k) {
      v2f a2;
      a2[0] = c.tscr[c.col * LD_T + 4 * k + c.rsel + 0];
      a2[1] = c.tscr[c.col * LD_T + 4 * k + c.rsel + 1];
      y = __builtin_amdgcn_wmma_f32_16x16x4_f32(
          false, a2, false, c.hB[k], (short)0, y, false, false);
    }

    // ---- scatter bands: one 64-bit base, then immediate-offset stores ----
    const int oy = oy0 + tr * 8, ox = ox0 + tc * 8;
    float* obase = outL + (size_t)(b * 4 + c.band) * ((size_t)W * W)
                        + (size_t)oy * W + (ox + c.cc);
#pragma unroll
    for (int r = 0; r < 8; ++r) obase[r * W] = y[r];

    if (HAS_NEXT && c.lane < 8) {
      float* nbase = nA + (tr * 8) * LDOUT + tc * 8 + c.cc;
#pragma unroll
      for (int r = 0; r < 8; ++r) nbase[r * LDOUT] = y[r];
    }
  }
}

__global__ __launch_bounds__(256) void haar3_wmma_kernel(
    const float* __restrict__ x, float* __restrict__ out) {
  __shared__ float sIn[64 * LD_IN];
  __shared__ float sL1[32 * LD_L1];
  __shared__ float sL2[16 * LD_L2];
  __shared__ float sT [8 * 16 * LD_T];

  const int tid  = threadIdx.x;
  const int lane = tid & 31;
  const int wave = tid >> 5;
  const int tX = blockIdx.x, tY = blockIdx.y, b = blockIdx.z;

  // ---- 64x64 tile of channel 0: global -> LDS ----
  const float* src = x + (size_t)b * 3u * 512u * 512u
                       + (size_t)(tY * 64) * 512u + (size_t)(tX * 64);
#if defined(__has_builtin) && __has_builtin(__builtin_amdgcn_global_load_async_to_lds_b128)
#pragma unroll
  for (int i = 0; i < 4; ++i) {
    const int linear = i * 256 + tid;
    const int row = linear >> 4;
    const int c4  = (linear & 15) << 2;
    v4i* g = (v4i*)(src + (size_t)row * 512 + c4);
    v4i* l = (v4i*)&sIn[row * LD_IN + c4];
    __builtin_amdgcn_global_load_async_to_lds_b128(g, l, 0, 0);
  }
  asm volatile("s_wait_asynccnt 0" ::: "memory");
#else
#pragma unroll
  for (int i = 0; i < 4; ++i) {
    const int linear = i * 256 + tid;
    const int row = linear >> 4;
    const int c4  = (linear & 15) << 2;
    const float4 v = *reinterpret_cast<const float4*>(src + (size_t)row * 512 + c4);
    *reinterpret_cast<float4*>(&sIn[row * LD_IN + c4]) = v;
  }
#endif
  __syncthreads();

  WaveCtx c;
  c.lane = lane; c.wave = wave;
  c.col  = lane & 15;
  c.rsel = (lane >= 16) ? 2 : 0;
  c.radd = (lane >= 16) ? 8 : 0;
  if      (lane <  8) { c.band = 0; c.cc = lane;      }   // cA
  else if (lane < 16) { c.band = 2; c.cc = lane - 8;  }   // cV
  else if (lane < 24) { c.band = 1; c.cc = lane - 16; }   // cH
  else                { c.band = 3; c.cc = lane - 24; }   // cD
#pragma unroll
  for (int k = 0; k < 4; ++k) {
    c.hA[k] = haar_op(k, lane, 0.5f);   // vertical stage carries the 0.5
    c.hB[k] = haar_op(k, lane, 1.0f);   // horizontal stage (H^T)
  }
  c.tscr = &sT[wave * 16 * LD_T];

  haar_level<64, 256, LD_IN, LD_L1, true >(sIn, sL1, out, b, tX, tY, c);
  __syncthreads();
  haar_level<32, 128, LD_L1, LD_L2, true >(sL1, sL2, out + 16777216ull, b, tX, tY, c);
  __syncthreads();
  haar_level<16,  64, LD_L2, 1,     false>(sL2, nullptr, out + 20971520ull, b, tX, tY, c);
}

extern "C" void kernel_launch(void* const* d_in, const int* in_sizes, int n_in,
                              void* d_out, int out_size, void* d_ws, size_t ws_size,
                              hipStream_t stream) {
  (void)in_sizes; (void)n_in; (void)out_size; (void)d_ws; (void)ws_size;
  const float* x = (const float*)d_in[0];
  float* out = (float*)d_out;
  dim3 grid(8, 8, 64), block(256);
  hipLaunchKernelGGL(haar3_wmma_kernel, grid, block, 0, stream, x, out);
}